// EdgeAttention_72456098284127
// MI455X (gfx1250) — compile-verified
//
#include <hip/hip_runtime.h>
#include <hip/hip_bf16.h>

// ---------------------------------------------------------------------------
// EdgeAttention on MI455X (gfx1250)
//   Pass 1: channel-mean via V_WMMA_F32_16X16X4_F32 (f32-exact), streams the
//           1 GiB input once (memory-bound roofline ~46us @ 23.3 TB/s).
//   Pass 2: fused LDS stencil chain on the 8 MB gray plane.
// ---------------------------------------------------------------------------

typedef __attribute__((ext_vector_type(2))) float v2f;
typedef __attribute__((ext_vector_type(8))) float v8f;

#define IMG_H 512
#define IMG_W 512
#define HW    (IMG_H * IMG_W)      // 262144 = 2^18
#define NCH   128
#define CHW   ((size_t)NCH * HW)
#define NB    8

// ---------------------------------------------------------------------------
// Pass 1: gray[b,h,w] = mean_c x[b,c,h,w]
// Each wave owns 64 consecutive pixels (4 tiles of 16) and accumulates
// 128 channels in K=4 steps with 4 independent WMMA chains.
// A = splat(1/128): because A is uniform over M and K, the reduction result
// is independent of the exact B-operand K-lane layout (sum is commutative).
// ---------------------------------------------------------------------------
__global__ __launch_bounds__(256) void gray_mean_wmma(
    const float* __restrict__ x, float* __restrict__ gray) {
  const int lane = threadIdx.x & 31;
  const int gw   = (blockIdx.x * blockDim.x + threadIdx.x) >> 5;  // global wave
  const size_t pixBase = (size_t)gw << 6;        // 64 pixels per wave
  const size_t b = pixBase >> 18;                // image index (HW = 2^18)
  const size_t p = pixBase & (size_t)(HW - 1);   // pixel within image
  const float* src = x + b * CHW + p;

  const int n  = lane & 15;   // pixel column within 16-tile
  const int hi = lane >> 4;   // lane half selects K={0,1} vs K={2,3} rows
  const size_t off0 = (size_t)(2 * hi) * HW + (size_t)n;

  const v2f a = {0.0078125f, 0.0078125f};        // 1/128 in both A VGPRs
  v8f c0 = {}, c1 = {}, c2 = {}, c3 = {};

#pragma unroll 4
  for (int cb = 0; cb < NCH; cb += 4) {
    const float* pc = src + (size_t)cb * HW;
    if (cb < NCH - 4) {  // speculative prefetch of next channel quad
      __builtin_prefetch((const void*)(pc + 4 * (size_t)HW + off0), 0, 1);
    }
    // B operand tiles: element0 -> VGPR0 (K=0/2 by lane half), element1 -> VGPR1 (K=1/3)
    v2f b0 = { pc[off0 +  0], pc[off0 +  0 + HW] };
    v2f b1 = { pc[off0 + 16], pc[off0 + 16 + HW] };
    v2f b2 = { pc[off0 + 32], pc[off0 + 32 + HW] };
    v2f b3 = { pc[off0 + 48], pc[off0 + 48 + HW] };
    c0 = __builtin_amdgcn_wmma_f32_16x16x4_f32(false, a, false, b0, (short)0, c0, false, false);
    c1 = __builtin_amdgcn_wmma_f32_16x16x4_f32(false, a, false, b1, (short)0, c1, false, false);
    c2 = __builtin_amdgcn_wmma_f32_16x16x4_f32(false, a, false, b2, (short)0, c2, false, false);
    c3 = __builtin_amdgcn_wmma_f32_16x16x4_f32(false, a, false, b3, (short)0, c3, false, false);
  }

  // D VGPR0: lanes 0-15 hold row M=0 (N=lane), lanes 16-31 hold row M=8
  // (N=lane-16). All rows are identical, so every lane's c[0] is the mean of
  // pixel (tileBase + (lane&15)). Two stores per lane cover 64 pixels.
  const float vA = hi ? c2[0] : c0[0];
  const float vB = hi ? c3[0] : c1[0];
  const size_t o = pixBase + (size_t)n + (size_t)(hi << 5);
  gray[o]      = vA;
  gray[o + 16] = vB;
}

// ---------------------------------------------------------------------------
// Pass 2: fused blur -> sobel -> |grad| -> depth 3x3 -> point 1x1 -> sigmoid
// 32x32 output tile, gray halo 4 (40x40), blurred 36x36, grad 34x34.
// Zero padding is applied independently at every conv stage (matches
// reference's per-conv explicit padding).
// ---------------------------------------------------------------------------
__global__ __launch_bounds__(256) void edge_pipeline(
    const float* __restrict__ gray,
    const float* __restrict__ depth_w, const float* __restrict__ depth_b,
    const float* __restrict__ point_w, const float* __restrict__ point_b,
    float* __restrict__ out) {
  // Normalized 5x5 Gaussian (sigma=1), by squared distance d=(dy-2)^2+(dx-2)^2
  const float w0 = 0.16210282f;   // d=0
  const float w1 = 0.09832033f;   // d=1
  const float w2 = 0.05963430f;   // d=2
  const float w4 = 0.02193825f;   // d=4
  const float w5 = 0.01330621f;   // d=5
  const float w8 = 0.00296901f;   // d=8
  const float GW[5][5] = {
    { w8, w5, w4, w5, w8 },
    { w5, w2, w1, w2, w5 },
    { w4, w1, w0, w1, w4 },
    { w5, w2, w1, w2, w5 },
    { w8, w5, w4, w5, w8 } };

  __shared__ float g[40][41];   // gray tile + halo 4   (odd stride: no bank conflicts)
  __shared__ float bl[36][37];  // blurred tile + halo 2
  __shared__ float gm[34][35];  // grad-mag tile + halo 1

  const int tx0 = blockIdx.x * 32;
  const int ty0 = blockIdx.y * 32;
  const int img = blockIdx.z;
  const int tid = threadIdx.x;
  const float* gp = gray + (size_t)img * HW;

  float dw[9];
#pragma unroll
  for (int i = 0; i < 9; ++i) dw[i] = depth_w[i];
  const float db = depth_b[0];
  const float pw = point_w[0];
  const float pb = point_b[0];

  // Stage A: load 40x40 gray tile (zero-pad outside image)
  for (int i = tid; i < 40 * 40; i += 256) {
    const int r = i / 40, cc = i - r * 40;
    const int y = ty0 + r - 4, xx = tx0 + cc - 4;
    float v = 0.0f;
    if ((unsigned)y < (unsigned)IMG_H && (unsigned)xx < (unsigned)IMG_W)
      v = gp[y * IMG_W + xx];
    g[r][cc] = v;
  }
  __syncthreads();

  // Stage B: 5x5 Gaussian on 36x36 region ([-2..33] rel. to tile)
  for (int i = tid; i < 36 * 36; i += 256) {
    const int r = i / 36, cc = i - r * 36;
    const int y = ty0 + r - 2, xx = tx0 + cc - 2;
    float v = 0.0f;
    if ((unsigned)y < (unsigned)IMG_H && (unsigned)xx < (unsigned)IMG_W) {
#pragma unroll
      for (int dy = 0; dy < 5; ++dy)
#pragma unroll
        for (int dx = 0; dx < 5; ++dx)
          v += GW[dy][dx] * g[r + dy][cc + dx];
    }
    bl[r][cc] = v;
  }
  __syncthreads();

  // Stage C: Sobel + magnitude on 34x34 region ([-1..32] rel. to tile)
  for (int i = tid; i < 34 * 34; i += 256) {
    const int r = i / 34, cc = i - r * 34;
    const int y = ty0 + r - 1, xx = tx0 + cc - 1;
    float v = 0.0f;
    if ((unsigned)y < (unsigned)IMG_H && (unsigned)xx < (unsigned)IMG_W) {
      const float a00 = bl[r    ][cc], a01 = bl[r    ][cc + 1], a02 = bl[r    ][cc + 2];
      const float a10 = bl[r + 1][cc],                           a12 = bl[r + 1][cc + 2];
      const float a20 = bl[r + 2][cc], a21 = bl[r + 2][cc + 1], a22 = bl[r + 2][cc + 2];
      const float gx = -a00 + a02 - 2.0f * a10 + 2.0f * a12 - a20 + a22;
      const float gy = -a00 - 2.0f * a01 - a02 + a20 + 2.0f * a21 + a22;
      v = sqrtf(gx * gx + gy * gy + 1e-8f);
    }
    gm[r][cc] = v;
  }
  __syncthreads();

  // Stage D: depth 3x3 + bias, point 1x1 + bias, sigmoid
  for (int i = tid; i < 32 * 32; i += 256) {
    const int r = i >> 5, cc = i & 31;
    float e = db;
#pragma unroll
    for (int dy = 0; dy < 3; ++dy)
#pragma unroll
      for (int dx = 0; dx < 3; ++dx)
        e += dw[dy * 3 + dx] * gm[r + dy][cc + dx];
    const float e2 = pw * e + pb;
    out[(size_t)img * HW + (size_t)(ty0 + r) * IMG_W + (tx0 + cc)] =
        1.0f / (1.0f + __expf(-e2));
  }
}

// ---------------------------------------------------------------------------
extern "C" void kernel_launch(void* const* d_in, const int* in_sizes, int n_in,
                              void* d_out, int out_size, void* d_ws, size_t ws_size,
                              hipStream_t stream) {
  const float* x       = (const float*)d_in[0];
  const float* depth_w = (const float*)d_in[1];
  const float* depth_b = (const float*)d_in[2];
  const float* point_w = (const float*)d_in[3];
  const float* point_b = (const float*)d_in[4];
  float* out  = (float*)d_out;
  float* gray = (float*)d_ws;   // needs 8*512*512*4 = 8 MB scratch

  // Pass 1: 8*512*512 pixels / 64 per wave = 32768 waves; 8 waves (256 thr) / block
  gray_mean_wmma<<<4096, 256, 0, stream>>>(x, gray);

  // Pass 2: 16x16 tiles per image, 8 images
  dim3 grid(IMG_W / 32, IMG_H / 32, NB);
  edge_pipeline<<<grid, 256, 0, stream>>>(gray, depth_w, depth_b,
                                          point_w, point_b, out);
}